// GCN_11836929868487
// MI455X (gfx1250) — compile-verified
//
#include <hip/hip_runtime.h>

typedef __attribute__((ext_vector_type(2))) float v2f;
typedef __attribute__((ext_vector_type(8))) float v8f;

#define NODES  50000
#define EDGES  800000
#define GRAPHS 64
#define C      64      // IN_C == HID_C
#define OUTC   32
#define LPITCH 68      // LDS pitch (floats) for transposed weights: conflict-free, 8B-aligned

// ---------------- elementwise helpers ----------------
__global__ void fill_kernel(float* __restrict__ p, float v, int n) {
  int i = blockIdx.x * blockDim.x + threadIdx.x;
  if (i < n) p[i] = v;
}

__global__ void deg_count_kernel(const int* __restrict__ dst, float* __restrict__ deg) {
  int i = blockIdx.x * blockDim.x + threadIdx.x;
  if (i < EDGES) atomicAdd(&deg[dst[i]], 1.0f);
}

__global__ void rsqrt_kernel(float* __restrict__ p, int n) {
  int i = blockIdx.x * blockDim.x + threadIdx.x;
  if (i < n) p[i] = rsqrtf(p[i]);
}

// ---------------- WMMA GEMM: out[r][:] = (X @ W)[r][:] * rowscale[r] ----------------
// X: [nrows x 64], W: [64 x 64] row-major, out: [nrows x 64]
// 8 waves / block, one 16-row tile per wave, full 64-wide N per wave (4 tiles).
// W is staged TRANSPOSED in LDS (pitch 68) so each B fragment is one aligned ds_load_b64.
__global__ __launch_bounds__(256)
void gemm_rs_wmma(const float* __restrict__ X, const float* __restrict__ W,
                  const float* __restrict__ rowscale, float* __restrict__ out,
                  int ntiles) {
  __shared__ float sWt[C * LPITCH];
  for (int i = threadIdx.x; i < C * C; i += 256) {
    const int k = i >> 6, n = i & 63;
    sWt[n * LPITCH + k] = W[i];
  }
  __syncthreads();

  const int wave = threadIdx.x >> 5;
  const int lane = threadIdx.x & 31;
  const int tile = blockIdx.x * 8 + wave;
  if (tile >= ntiles) return;

  const int m0 = tile * 16;
  const int ar = m0 + (lane & 15);   // A-matrix row held by this lane
  const int kg = (lane >> 4) * 2;    // K offset of this lane group (always even)
  const int cn = lane & 15;          // B/D column within an N-tile
  const float* bp = sWt + cn * LPITCH;

  v8f acc0 = {}, acc1 = {}, acc2 = {}, acc3 = {};
#pragma unroll
  for (int ks = 0; ks < 16; ++ks) {
    const int k0 = ks * 4 + kg;
    v2f a = *(const v2f*)(X + ar * C + k0);               // global_load_b64
    v2f b0 = *(const v2f*)(bp + k0);                      // ds_load_b64 (aligned pair)
    v2f b1 = *(const v2f*)(bp + 16 * LPITCH + k0);
    v2f b2 = *(const v2f*)(bp + 32 * LPITCH + k0);
    v2f b3 = *(const v2f*)(bp + 48 * LPITCH + k0);
    acc0 = __builtin_amdgcn_wmma_f32_16x16x4_f32(false, a, false, b0, (short)0, acc0, false, false);
    acc1 = __builtin_amdgcn_wmma_f32_16x16x4_f32(false, a, false, b1, (short)0, acc1, false, false);
    acc2 = __builtin_amdgcn_wmma_f32_16x16x4_f32(false, a, false, b2, (short)0, acc2, false, false);
    acc3 = __builtin_amdgcn_wmma_f32_16x16x4_f32(false, a, false, b3, (short)0, acc3, false, false);
  }

#pragma unroll
  for (int r = 0; r < 8; ++r) {
    const int row = m0 + r + 8 * (lane >> 4);   // D layout: VGPR r -> row r (+8 for hi lanes)
    const float s = rowscale[row];
    out[row * C +  0 + cn] = acc0[r] * s;
    out[row * C + 16 + cn] = acc1[r] * s;
    out[row * C + 32 + cn] = acc2[r] * s;
    out[row * C + 48 + cn] = acc3[r] * s;
  }
}

// ---------------- edge scatter-add: acc[dst] += hs[src] ----------------
__global__ void scatter_kernel(const int* __restrict__ src, const int* __restrict__ dst,
                               const float* __restrict__ hs, float* __restrict__ acc) {
  int idx = blockIdx.x * blockDim.x + threadIdx.x;   // EDGES*16 threads
  int e  = idx >> 4;
  int cg = (idx & 15) << 2;
  if (e < EDGES) {
    int s = src[e], d = dst[e];
    const float4 v = *(const float4*)(hs + (size_t)s * C + cg);
    float* a = acc + (size_t)d * C + cg;
    atomicAdd(a + 0, v.x);
    atomicAdd(a + 1, v.y);
    atomicAdd(a + 2, v.z);
    atomicAdd(a + 3, v.w);
  }
}

// ---------------- finalize: acc = [relu]( dinv[i]*(acc + hs) + b[c] ) in place ----------------
__global__ void finalize_kernel(float* __restrict__ acc, const float* __restrict__ hs,
                                const float* __restrict__ dinv, const float* __restrict__ bias,
                                int do_relu) {
  int i = blockIdx.x * blockDim.x + threadIdx.x;     // NODES*C threads
  if (i < NODES * C) {
    int row = i >> 6, c = i & 63;
    float v = dinv[row] * (acc[i] + hs[i]) + bias[c];
    if (do_relu) v = fmaxf(v, 0.0f);
    acc[i] = v;
  }
}

// ---------------- global add pool: u[batch[i]] += z[i] ----------------
__global__ void pool_kernel(const float* __restrict__ z, const int* __restrict__ batch,
                            float* __restrict__ u) {
  int idx = blockIdx.x * blockDim.x + threadIdx.x;   // NODES*16 threads
  int i  = idx >> 4;
  int cg = (idx & 15) << 2;
  if (i < NODES) {
    int g = batch[i];
    const float4 v = *(const float4*)(z + (size_t)i * C + cg);
    float* p = u + g * C + cg;
    atomicAdd(p + 0, v.x);
    atomicAdd(p + 1, v.y);
    atomicAdd(p + 2, v.z);
    atomicAdd(p + 3, v.w);
  }
}

// ---------------- head: y = u @ Wf + bf  (64x64 @ 64x32), plus stream u to output ----------------
__global__ __launch_bounds__(256)
void head_wmma(const float* __restrict__ u, const float* __restrict__ Wf,
               const float* __restrict__ bf, float* __restrict__ y,
               float* __restrict__ out_u) {
  __shared__ float sWf[OUTC * LPITCH];                 // Wf transposed, pitch 68
  for (int i = threadIdx.x; i < C * OUTC; i += 256) {
    const int k = i >> 5, n = i & 31;
    sWf[n * LPITCH + k] = Wf[i];
  }
  __syncthreads();

  const int wave = threadIdx.x >> 5;
  const int lane = threadIdx.x & 31;
  const int mt = wave >> 1;                 // 4 M-tiles
  const int nt = wave & 1;                  // 2 N-tiles
  const int ar = mt * 16 + (lane & 15);
  const int kg = (lane >> 4) * 2;
  const int cn = nt * 16 + (lane & 15);
  const float* bp = sWf + cn * LPITCH;

  v8f acc = {};
#pragma unroll
  for (int ks = 0; ks < 16; ++ks) {
    const int k0 = ks * 4 + kg;
    v2f a = *(const v2f*)(u + ar * C + k0);
    v2f b = *(const v2f*)(bp + k0);
    acc = __builtin_amdgcn_wmma_f32_16x16x4_f32(false, a, false, b, (short)0, acc, false, false);
  }
#pragma unroll
  for (int r = 0; r < 8; ++r) {
    const int row = mt * 16 + r + 8 * (lane >> 4);
    y[row * OUTC + cn] = acc[r] + bf[cn];
  }
  // second tuple output: copy u (64*64 = 4096 = 16*256 elements)
#pragma unroll
  for (int j = 0; j < 16; ++j) {
    const int idx = threadIdx.x + j * 256;
    out_u[idx] = u[idx];
  }
}

extern "C" void kernel_launch(void* const* d_in, const int* in_sizes, int n_in,
                              void* d_out, int out_size, void* d_ws, size_t ws_size,
                              hipStream_t stream) {
  (void)in_sizes; (void)n_in; (void)out_size; (void)ws_size;

  const float* x     = (const float*)d_in[0];
  const int*   ei    = (const int*)  d_in[1];
  const int*   batch = (const int*)  d_in[2];
  const float* W1    = (const float*)d_in[3];
  const float* b1    = (const float*)d_in[4];
  const float* W2    = (const float*)d_in[5];
  const float* b2    = (const float*)d_in[6];
  const float* Wf    = (const float*)d_in[7];
  const float* bf    = (const float*)d_in[8];
  float* out = (float*)d_out;

  const int* src = ei;
  const int* dst = ei + EDGES;

  // workspace layout (floats): dinv | hs | acc | u   (~25.8 MB total)
  float* ws   = (float*)d_ws;
  float* dinv = ws;                       // NODES (padded to 50048 for alignment)
  float* hs   = ws + 50048;               // NODES*C
  float* acc  = hs + (size_t)NODES * C;   // NODES*C
  float* u    = acc + (size_t)NODES * C;  // GRAPHS*C

  const dim3 B(256);
  const int NT = NODES / 16;              // 3125 row tiles (exact)
  const int GB = (NT + 7) / 8;

  // degree (self-loops folded in as deg=1 init), then dinv = rsqrt(deg) in place
  fill_kernel     <<<(NODES + 255) / 256, B, 0, stream>>>(dinv, 1.0f, NODES);
  deg_count_kernel<<<(EDGES + 255) / 256, B, 0, stream>>>(dst, dinv);
  rsqrt_kernel    <<<(NODES + 255) / 256, B, 0, stream>>>(dinv, NODES);

  // ---- layer 1: hs = (x @ W1) * dinv ; acc = scatter(hs) ; acc = relu(dinv*(acc+hs)+b1)
  fill_kernel    <<<(NODES * C) / 256, B, 0, stream>>>(acc, 0.0f, NODES * C);
  gemm_rs_wmma   <<<GB, B, 0, stream>>>(x, W1, dinv, hs, NT);
  scatter_kernel <<<(EDGES * 16) / 256, B, 0, stream>>>(src, dst, hs, acc);
  finalize_kernel<<<(NODES * C) / 256, B, 0, stream>>>(acc, hs, dinv, b1, 1);

  // ---- layer 2: hs = (acc @ W2) * dinv ; acc = scatter(hs) ; acc = dinv*(acc+hs)+b2
  gemm_rs_wmma   <<<GB, B, 0, stream>>>(acc, W2, dinv, hs, NT);
  fill_kernel    <<<(NODES * C) / 256, B, 0, stream>>>(acc, 0.0f, NODES * C);
  scatter_kernel <<<(EDGES * 16) / 256, B, 0, stream>>>(src, dst, hs, acc);
  finalize_kernel<<<(NODES * C) / 256, B, 0, stream>>>(acc, hs, dinv, b2, 0);

  // ---- pool + head
  fill_kernel<<<(GRAPHS * C + 255) / 256, B, 0, stream>>>(u, 0.0f, GRAPHS * C);
  pool_kernel<<<(NODES * 16) / 256, B, 0, stream>>>(acc, batch, u);
  head_wmma  <<<1, B, 0, stream>>>(u, Wf, bf, out, out + GRAPHS * OUTC);
}